// GeneralizedFiringRateModel_84550726189065
// MI455X (gfx1250) — compile-verified
//
#include <hip/hip_runtime.h>
#include <hip/hip_bf16.h>
#include <math.h>

typedef __attribute__((ext_vector_type(2))) float v2f;
typedef __attribute__((ext_vector_type(8))) float v8f;

#define T_DIM 2000
#define B_DIM 2048
#define K_LAG 64

// ---------------------------------------------------------------------------
// Kernel 1: FIR part  x[t,b] = sum_k a[k] * currents[t-63+k, b]
// Banded-Toeplitz matmul via V_WMMA_F32_16X16X4_F32 (full fp32 precision).
// One wave computes a 16(t) x 64(b) tile: 4 accumulators share one A operand
// per k-chunk (20 chunks, fully unrolled -> 80 WMMAs). All operand masking is
// branchless (padded LDS band for A, clamped row + cndmask for B), so EXEC
// stays all-ones throughout. Result written into d_out (overwritten later).
// ---------------------------------------------------------------------------
__global__ __launch_bounds__(128) void fir_wmma_kernel(
    const float* __restrict__ currents,   // [T, B]
    const float* __restrict__ a,          // [64]
    float* __restrict__ xout)             // [T, B] (d_out, fully overwritten)
{
    // Padded Toeplitz band: aPad[16 + i] = a[i], zeros in the guard zones.
    __shared__ float aPad[96];
    const int tid = threadIdx.x;
    if (tid < 96) aPad[tid] = 0.f;
    __syncthreads();
    if (tid < K_LAG) aPad[16 + tid] = a[tid];
    __syncthreads();

    const int wave   = tid >> 5;               // 0..3 (wave-uniform)
    const int lane   = tid & 31;
    const int tileId = blockIdx.x * 4 + wave;  // 0..3999 (125 x 32 tiles)
    const int tileT  = tileId >> 5;            // /32
    const int tileB  = tileId & 31;            // %32
    if (tileT >= (T_DIM / 16)) return;         // wave-uniform; never taken

    const int t0 = tileT * 16;
    const int b0 = tileB * 64;

    const int half = lane >> 4;    // 0 or 1 : selects K-pair
    const int mn   = lane & 15;    // M-row for A operand, N-col for B operand
    const int kp   = half * 2;     // K base within chunk (0 or 2)

    v8f acc0 = {0.f,0.f,0.f,0.f,0.f,0.f,0.f,0.f};
    v8f acc1 = acc0, acc2 = acc0, acc3 = acc0;

    // Contract over global time index u in [t0-63, t0+16], 20 chunks of 4.
    //   A[m][j] = a[4c + kp + j - m]          (banded; guard zones are zero)
    //   B[j][n] = currents[u0 + kp + j, col]  (clamped row, cndmask to zero)
    #pragma unroll
    for (int c = 0; c < 20; ++c) {
        const int ia = 4 * c + kp - mn;        // in [-15, 78]
        v2f Aop;
        Aop.x = aPad[16 + ia];
        Aop.y = aPad[17 + ia];

        const int ua  = t0 - 63 + 4 * c + kp;
        const int ub  = ua + 1;
        const int uac = ua < 0 ? 0 : (ua >= T_DIM ? T_DIM - 1 : ua);
        const int ubc = ub < 0 ? 0 : (ub >= T_DIM ? T_DIM - 1 : ub);
        const bool va = (ua >= 0) && (ua < T_DIM);
        const bool vb = (ub >= 0) && (ub < T_DIM);
        const float* rowA = currents + uac * B_DIM + b0 + mn;
        const float* rowB = currents + ubc * B_DIM + b0 + mn;

        v2f B0, B1, B2, B3;
        B0.x = rowA[ 0]; B0.y = rowB[ 0];
        B1.x = rowA[16]; B1.y = rowB[16];
        B2.x = rowA[32]; B2.y = rowB[32];
        B3.x = rowA[48]; B3.y = rowB[48];
        if (!va) { B0.x = 0.f; B1.x = 0.f; B2.x = 0.f; B3.x = 0.f; }
        if (!vb) { B0.y = 0.f; B1.y = 0.f; B2.y = 0.f; B3.y = 0.f; }

        acc0 = __builtin_amdgcn_wmma_f32_16x16x4_f32(false, Aop, false, B0,
                                                     (short)0, acc0, false, false);
        acc1 = __builtin_amdgcn_wmma_f32_16x16x4_f32(false, Aop, false, B1,
                                                     (short)0, acc1, false, false);
        acc2 = __builtin_amdgcn_wmma_f32_16x16x4_f32(false, Aop, false, B2,
                                                     (short)0, acc2, false, false);
        acc3 = __builtin_amdgcn_wmma_f32_16x16x4_f32(false, Aop, false, B3,
                                                     (short)0, acc3, false, false);
    }

    // C/D layout: lane -> N = mn ; VGPR r -> M = r + 8*half
    #pragma unroll
    for (int r = 0; r < 8; ++r) {
        const int m = r + 8 * half;
        float* orow = xout + (t0 + m) * B_DIM + b0 + mn;
        orow[ 0] = acc0[r];
        orow[16] = acc1[r];
        orow[32] = acc2[r];
        orow[48] = acc3[r];
    }
}

// ---------------------------------------------------------------------------
// Kernel 2: the true recurrence.  One thread per batch element.
//   y[t] = sum_l (1000*b_lag[l]) * f[t-64+l]
//   f[t] = relu(mfr * tanh(poly((x[t]+y[t]-b_act)/mc)))
// f-history per lane in LDS, double-written ring hist[128][64]: each step's
// 64 taps are contiguous slots -> ds_load_b32 with immediate offsets,
// [slot][lane] layout is bank-conflict-free. x[t+1] is software-pipelined.
// ---------------------------------------------------------------------------
__global__ __launch_bounds__(64) void recurrence_kernel(
    const float* __restrict__ b_lag,       // [64]
    const float* __restrict__ poly_coeff,  // [1,4]
    const float* __restrict__ b_act,       // [1]
    const float* __restrict__ max_current, // [1]
    const float* __restrict__ max_fr,      // [1]
    float* __restrict__ out)               // [T,B]: holds x on entry, f on exit
{
    __shared__ float hist[128 * 64];       // 32 KB
    const int tid = threadIdx.x;           // 0..63
    const int b   = blockIdx.x * 64 + tid; // 0..2047

    float bl[K_LAG];
    #pragma unroll
    for (int l = 0; l < K_LAG; ++l) bl[l] = 1000.0f * b_lag[l];

    const float c0 = poly_coeff[0] * poly_coeff[0];
    const float c1 = poly_coeff[1] * poly_coeff[1];
    const float c2 = poly_coeff[2] * poly_coeff[2];
    const float c3 = poly_coeff[3] * poly_coeff[3];
    const float ba     = b_act[0];
    const float inv_mc = 1.0f / max_current[0];
    const float mfr    = max_fr[0];

    #pragma unroll
    for (int s = 0; s < 128; ++s) hist[s * 64 + tid] = 0.f;
    __syncthreads();   // each lane touches only its own column; kept for safety

    float xv = out[b];                      // x[0, b]
    for (int t = 0; t < T_DIM; ++t) {
        // Issue next step's x load early so it hides under the LDS dot.
        const int tn = (t + 1 < T_DIM) ? (t + 1) : (T_DIM - 1);
        const float xnext = out[tn * B_DIM + b];

        const int base = t & 63;
        const float* hcol = &hist[base * 64 + tid];

        float y0 = 0.f, y1 = 0.f, y2 = 0.f, y3 = 0.f;
        #pragma unroll
        for (int l = 0; l < K_LAG; l += 4) {
            y0 = fmaf(bl[l + 0], hcol[(l + 0) * 64], y0);
            y1 = fmaf(bl[l + 1], hcol[(l + 1) * 64], y1);
            y2 = fmaf(bl[l + 2], hcol[(l + 2) * 64], y2);
            y3 = fmaf(bl[l + 3], hcol[(l + 3) * 64], y3);
        }
        const float y = (y0 + y1) + (y2 + y3);

        const float z  = xv + y;
        const float xn = (z - ba) * inv_mc;
        const float p  = fmaf(xn, fmaf(xn, fmaf(xn, c3, c2), c1), c0);
        float f = mfr * tanhf(p);
        f = f > 0.f ? f : 0.f;

        out[t * B_DIM + b] = f;                        // overwrite x with f
        hist[base * 64 + tid]        = f;              // ring slot
        hist[(base + 64) * 64 + tid] = f;              // duplicate for contiguity
        xv = xnext;
    }
}

extern "C" void kernel_launch(void* const* d_in, const int* in_sizes, int n_in,
                              void* d_out, int out_size, void* d_ws, size_t ws_size,
                              hipStream_t stream) {
    const float* currents    = (const float*)d_in[0];  // [2000,2048]
    const float* a           = (const float*)d_in[1];  // [64]
    const float* b_lag       = (const float*)d_in[2];  // [64]
    const float* poly_coeff  = (const float*)d_in[3];  // [1,4]
    const float* b_act       = (const float*)d_in[4];  // [1]
    const float* max_current = (const float*)d_in[5];  // [1]
    const float* max_fr      = (const float*)d_in[6];  // [1]
    float* out = (float*)d_out;                        // [2000,2048]

    // 125 t-tiles x 32 b-tiles = 4000 wave-tiles, 4 waves/block -> 1000 blocks.
    fir_wmma_kernel<<<1000, 128, 0, stream>>>(currents, a, out);
    // 2048 batch lanes, 64 threads (2 waves) per block -> 32 blocks.
    recurrence_kernel<<<32, 64, 0, stream>>>(b_lag, poly_coeff, b_act,
                                             max_current, max_fr, out);
}